// Dcls1d_12713103196284
// MI455X (gfx1250) — compile-verified
//
#include <hip/hip_runtime.h>
#include <cstdint>

// Dcls1d: depthwise dilated conv with learnable fractional tap positions.
// x[16,256,8192] f32, weight[256,1,7], P[1,256,1,7], bias[256] -> out[16,256,8191]
// Memory-bound (268 MB @ 23.3 TB/s ~ 11.5us). Strategy: async global->LDS tile
// staging (CDNA5 ASYNCcnt path), 14-tap sparse stencil from registers.

#define L_IN   8192
#define L_OUT  8191   // 8192 + 2*27 - 56 + 1
#define CCH    256
#define NBATCH 16
#define KS     7
#define TILE   2048
#define SPAN   (TILE + 56)   // 2104 floats staged per block (halo 28 each side)
#define TPB    256

#if defined(__gfx1250__) && __has_builtin(__builtin_amdgcn_global_load_async_to_lds_b128)
#define USE_ASYNC_LDS 1
#warning "probe: __builtin_amdgcn_global_load_async_to_lds_b128 AVAILABLE (async path)"
#else
#define USE_ASYNC_LDS 0
#warning "probe: async-to-lds builtin NOT available on this pass (fallback staging)"
#endif

#if USE_ASYNC_LDS
// Builtin signature (from clang diagnostic): param0 is
// 'int __attribute__((vector_size(16))) __device__ *'  i.e. v4i addrspace(1)*.
typedef int v4i __attribute__((__vector_size__(16)));
typedef __attribute__((address_space(1))) v4i as1_v4i;
typedef __attribute__((address_space(3))) v4i as3_v4i;

__device__ __forceinline__ void wait_async0() {
#if __has_builtin(__builtin_amdgcn_s_wait_asynccnt)
  __builtin_amdgcn_s_wait_asynccnt(0);
#else
  asm volatile("s_wait_asynccnt 0" ::: "memory");
#endif
}
#endif

__global__ __launch_bounds__(TPB) void dcls1d_kernel(
    const float* __restrict__ x, const float* __restrict__ weight,
    const float* __restrict__ P, const float* __restrict__ bias,
    float* __restrict__ out)
{
  __shared__ __align__(16) float tile[SPAN];

  const int tid = threadIdx.x;
  const int t0  = blockIdx.x * TILE;   // first output index of this tile
  const int c   = blockIdx.y;
  const int n   = blockIdx.z;

  const float* xrow = x   + ((size_t)n * CCH + c) * L_IN;
  float*       orow = out + ((size_t)n * CCH + c) * L_OUT;

  // ---- per-channel sparse taps (uniform across block; recomputed in regs) ----
  int   toff[2 * KS];
  float twt [2 * KS];
  {
    const float* wrow = weight + c * KS;
    const float* prow = P      + c * KS;
#pragma unroll
    for (int k = 0; k < KS; ++k) {
      float w = wrow[k];
      float p = prow[k] + (float)(k * 8 + 4);      // nominal center k*dil + dil/2
      p = fminf(fmaxf(p, 0.0f), 55.0f);
      float fi = floorf(p);
      float r  = p - fi;
      int i0 = (int)fi;
      int i1 = i0 + 1; if (i1 > 55) i1 = 55;
      toff[2 * k]     = i0; twt[2 * k]     = w * (1.0f - r);
      toff[2 * k + 1] = i1; twt[2 * k + 1] = w * r;
    }
  }

  // ---- stage x[g_start .. g_start+SPAN) into LDS (zero pad outside row) ----
  const int g_start = t0 - 28;                      // multiple of 4 -> 16B aligned
  const int lo = (g_start < 0) ? 0 : g_start;
  int hi = g_start + SPAN; if (hi > L_IN) hi = L_IN; // lo/hi are multiples of 4

  for (int i = tid; i < SPAN; i += TPB) {           // zero-fill out-of-row halo
    int g = g_start + i;
    if (g < lo || g >= hi) tile[i] = 0.0f;
  }

  const int nvec = (hi - lo) >> 2;                  // float4 chunks
  const float* gsrc = xrow + lo;
  float* ldst = &tile[lo - g_start];

#if USE_ASYNC_LDS
  for (int v = tid; v < nvec; v += TPB) {
    as1_v4i* gp = (as1_v4i*)(uintptr_t)(gsrc + 4 * v);
    // low 32 bits of a flat LDS address are the LDS byte offset (AS3 ptr value)
    as3_v4i* lp = (as3_v4i*)(uint32_t)(uintptr_t)(ldst + 4 * v);
    __builtin_amdgcn_global_load_async_to_lds_b128(gp, lp, 0, 0);
  }
  wait_async0();           // each wave drains its own ASYNCcnt before barrier
#else
  for (int v = tid; v < nvec; v += TPB) {
    ((float4*)ldst)[v] = ((const float4*)gsrc)[v];
  }
#endif
  __syncthreads();

  // ---- 14-tap stencil: out[l] = bias + sum_j w_j * x[l - 27 + off_j] ----
  // lds index = (t0+t) - 27 + off - g_start = t + 1 + off   (in [1, 2103])
  const float b = bias[c];
#pragma unroll
  for (int i = 0; i < TILE / TPB; ++i) {
    const int t = i * TPB + tid;
    float acc = b;
#pragma unroll
    for (int j = 0; j < 2 * KS; ++j)
      acc = fmaf(twt[j], tile[t + 1 + toff[j]], acc);
    if (t0 + t < L_OUT) orow[t0 + t] = acc;   // coalesced b32 stores
  }
}

extern "C" void kernel_launch(void* const* d_in, const int* in_sizes, int n_in,
                              void* d_out, int out_size, void* d_ws, size_t ws_size,
                              hipStream_t stream) {
  (void)in_sizes; (void)n_in; (void)d_ws; (void)ws_size; (void)out_size;
  const float* x      = (const float*)d_in[0];
  const float* weight = (const float*)d_in[1];
  const float* P      = (const float*)d_in[2];
  const float* bias   = (const float*)d_in[3];
  float* out = (float*)d_out;

  dim3 grid(L_IN / TILE, CCH, NBATCH);   // 4 x 256 x 16 blocks
  dcls1d_kernel<<<grid, TPB, 0, stream>>>(x, weight, P, bias, out);
}